// PerturbedTopK_40965398069592
// MI455X (gfx1250) — compile-verified
//
#include <hip/hip_runtime.h>
#include <stdint.h>

// Problem constants (from the reference setup)
#define Bdim   64
#define Nn     1000
#define Dd     1024
#define Kk     256
#define SIGMA  0.05f
#define INV_N  (1.0f / 1000.0f)

#define WPB    4                          // waves per block (128 threads)
#define RPW    4                          // rows per wave (1000 % 4 == 0 -> same b per chunk)
#define NBLK   ((Bdim * Nn) / (WPB * RPW))   // 4000 blocks

// Issue one 4KB noise row as 8 async b128 global->LDS DMA transfers.
// lane L moves 16B at byte offset 16*L + 512*t; ASYNCcnt += 8.
__device__ __forceinline__ void async_row_to_lds(uint32_t lds_byte, uint64_t gbase, int lane) {
  uint32_t voff = (uint32_t)lane * 16u;
#pragma unroll
  for (int t = 0; t < 8; ++t) {
    uint32_t l = lds_byte + voff + (uint32_t)t * 512u;
    uint32_t g = voff + (uint32_t)t * 512u;
    asm volatile("global_load_async_to_lds_b128 %0, %1, %2"
                 :: "v"(l), "v"(g), "s"(gbase)
                 : "memory");
  }
}

__global__ __launch_bounds__(WPB * 32)
void ptopk_kernel(const float* __restrict__ x,
                  const float* __restrict__ noise,
                  float* __restrict__ out) {
  __shared__ float lds[WPB][2][Dd];   // double buffer, 4KB per buffer per wave (32KB total)

  const int lane = threadIdx.x & 31;
  const int wave = threadIdx.x >> 5;
  const int gw   = blockIdx.x * WPB + wave;
  const int row0 = gw * RPW;          // rows [row0, row0+RPW) share the same b
  const int b    = row0 / Nn;

  // x[b, j] for this lane's interleaved elements j = lane + 32t (L2-hot, 256KB total)
  float xv[32];
#pragma unroll
  for (int t = 0; t < 32; ++t) xv[t] = x[b * Dd + t * 32 + lane];

  const uint32_t lds0 = (uint32_t)(uintptr_t)(&lds[wave][0][0]);
  const uint32_t lds1 = (uint32_t)(uintptr_t)(&lds[wave][1][0]);
  const uint64_t outbase = (uint64_t)(uintptr_t)out;

  // Prologue: DMA row0 into buffer 0
  async_row_to_lds(lds0, (uint64_t)(uintptr_t)(noise + (size_t)row0 * Dd), lane);

  for (int i = 0; i < RPW; ++i) {
    // Prefetch next row into the other buffer while we process this one
    if (i + 1 < RPW) {
      async_row_to_lds((i & 1) ? lds0 : lds1,
                       (uint64_t)(uintptr_t)(noise + (size_t)(row0 + i + 1) * Dd), lane);
      // 16 outstanding max; async loads complete in order, so <=8 means row i landed
      asm volatile("s_wait_asynccnt 0x8" ::: "memory");
    } else {
      asm volatile("s_wait_asynccnt 0x0" ::: "memory");
    }

    const float* rl = &lds[wave][i & 1][0];

    // Order-preserving uint keys: unsigned compare == float compare
    uint32_t u[32];
#pragma unroll
    for (int t = 0; t < 32; ++t) {
      float p = fmaf(SIGMA, rl[t * 32 + lane], xv[t]);  // conflict-free ds_load_b32
      uint32_t bs = __float_as_uint(p);
      u[t] = bs ^ ((uint32_t)((int32_t)bs >> 31) | 0x80000000u);
    }

    // Radix binary search for the Kk-th largest key:
    // maximal T with count(u >= T) >= Kk  ->  T == Kk-th largest (distinct values)
    uint32_t T = 0;
    for (int bit = 31; bit >= 0; --bit) {
      uint32_t cand = T | (1u << bit);
      int c = 0;
#pragma unroll
      for (int t = 0; t < 32; ++t) c += (u[t] >= cand) ? 1 : 0;
#pragma unroll
      for (int off = 16; off; off >>= 1) c += __shfl_xor(c, off, 32);
      if (c >= Kk) T = cand;
    }

    // Scatter: slot kk = number of selected indices < j (== position after jnp.sort(idx)).
    // Block t covers contiguous indices [32t, 32t+32): ballot gives the in-block ranks.
    const float addval = INV_N;
    int running = 0;
#pragma unroll
    for (int t = 0; t < 32; ++t) {
      bool sel = (u[t] >= T);
      uint32_t bal = __builtin_amdgcn_ballot_w32(sel);
      if (sel) {
        int kk = running + __popc(bal & ((1u << lane) - 1u));
        if (kk < Kk) {  // tie insurance
          int j = t * 32 + lane;
          uint32_t boff = ((((uint32_t)(b * Kk + kk)) << 10) + (uint32_t)j) << 2;
          // Guaranteed HW float atomic (no-return form, STOREcnt-tracked);
          // s_endpgm's implicit wait-idle covers completion.
          asm volatile("global_atomic_add_f32 %0, %1, %2"
                       :: "v"(boff), "v"(addval), "s"(outbase)
                       : "memory");
        }
      }
      running += __popc(bal);
    }
  }
}

extern "C" void kernel_launch(void* const* d_in, const int* in_sizes, int n_in,
                              void* d_out, int out_size, void* d_ws, size_t ws_size,
                              hipStream_t stream) {
  (void)in_sizes; (void)n_in; (void)d_ws; (void)ws_size;
  const float* x     = (const float*)d_in[0];
  const float* noise = (const float*)d_in[1];
  // d_in[2] is k == 256; compile-time constant Kk matches the reference setup.
  float* out = (float*)d_out;

  // Output accumulates via atomics -> zero it first (capture-safe async memset)
  hipMemsetAsync(d_out, 0, (size_t)out_size * sizeof(float), stream);

  ptopk_kernel<<<NBLK, WPB * 32, 0, stream>>>(x, noise, out);
}